// SelfReferentialTitans_34041910788569
// MI455X (gfx1250) — compile-verified
//
#include <hip/hip_runtime.h>
#include <math.h>

#define NH   12
#define DD   64
#define DIMM 768
#define NB   4
#define SS   2048
#define BS   (NB*SS)           // 8192 rows
#define RSZ  ((size_t)BS*DIMM) // 6291456 floats per [B*S, DIM] matrix

typedef __attribute__((ext_vector_type(2))) float v2f;
typedef __attribute__((ext_vector_type(8))) float v8f;

__device__ __forceinline__ float sigmoidf_(float x) { return 1.0f / (1.0f + expf(-x)); }
__device__ __forceinline__ float geluf_(float x)    { return 0.5f * x * (1.0f + erff(x * 0.70710678118654752f)); }

// ---------------------------------------------------------------------------
// 1) Input LayerNorm over DIM: one block per row of [B*S, DIM]
// ---------------------------------------------------------------------------
__global__ __launch_bounds__(256) void ln_in_kernel(const float* __restrict__ x,
                                                    const float* __restrict__ g,
                                                    const float* __restrict__ b,
                                                    float* __restrict__ xn) {
    const int tid = threadIdx.x;
    const size_t row = blockIdx.x;
    const float* xr = x + row * DIMM;
    float v0 = xr[tid], v1 = xr[tid + 256], v2 = xr[tid + 512];
    float s  = v0 + v1 + v2;
    float s2 = v0 * v0 + v1 * v1 + v2 * v2;
    for (int o = 16; o >= 1; o >>= 1) { s += __shfl_xor(s, o, 32); s2 += __shfl_xor(s2, o, 32); }
    __shared__ float ps[8], ps2[8];
    const int wid = tid >> 5;
    if ((tid & 31) == 0) { ps[wid] = s; ps2[wid] = s2; }
    __syncthreads();
    float ts = 0.f, ts2 = 0.f;
    for (int i = 0; i < 8; ++i) { ts += ps[i]; ts2 += ps2[i]; }
    const float mean = ts * (1.0f / DIMM);
    const float var  = ts2 * (1.0f / DIMM) - mean * mean;
    const float inv  = rsqrtf(var + 1e-5f);
    float* xo = xn + row * DIMM;
    xo[tid]       = (v0 - mean) * inv * g[tid]       + b[tid];
    xo[tid + 256] = (v1 - mean) * inv * g[tid + 256] + b[tid + 256];
    xo[tid + 512] = (v2 - mean) * inv * g[tid + 512] + b[tid + 512];
}

// ---------------------------------------------------------------------------
// 2) Generic [8192x768] @ [768x768] GEMM, fp32 WMMA 16x16x4.
//    One wave per block; each wave owns a 64x64 output tile (16 accumulators),
//    reusing each A fragment across 4 N-tiles and each B fragment across 4
//    M-tiles: 8 fragment loads per 16 WMMAs. Grid: (8192/64, 768/64) = (128, 12).
// ---------------------------------------------------------------------------
__global__ __launch_bounds__(32) void gemm768_kernel(const float* __restrict__ A,
                                                     const float* __restrict__ W,
                                                     float* __restrict__ out) {
    const int lane = threadIdx.x;
    const int m0 = blockIdx.x * 64;
    const int n0 = blockIdx.y * 64;
    const int r  = lane & 15;            // row-in-tile (A) / col-in-tile (B,D)
    const int kb = (lane >> 4) * 2;      // K sub-offset 0 or 2

    const float* arow0 = A + (size_t)(m0 + r) * DIMM;
    const float* wcol  = W + n0 + r;

    v8f acc[4][4] = {};
    for (int k = 0; k < DIMM; k += 4) {
        v2f a[4], bm[4];
#pragma unroll
        for (int i = 0; i < 4; ++i) {
            const float* ar = arow0 + (size_t)i * 16 * DIMM;
            a[i][0] = ar[k + kb];
            a[i][1] = ar[k + kb + 1];
        }
#pragma unroll
        for (int j = 0; j < 4; ++j) {
            bm[j][0] = wcol[(size_t)(k + kb) * DIMM + j * 16];
            bm[j][1] = wcol[(size_t)(k + kb + 1) * DIMM + j * 16];
        }
#pragma unroll
        for (int i = 0; i < 4; ++i)
#pragma unroll
            for (int j = 0; j < 4; ++j)
                acc[i][j] = __builtin_amdgcn_wmma_f32_16x16x4_f32(false, a[i], false, bm[j],
                                                                 (short)0, acc[i][j], false, false);
    }
    const int rowadd = (lane >> 4) * 8;   // lanes 16-31 hold rows +8
#pragma unroll
    for (int i = 0; i < 4; ++i) {
#pragma unroll
        for (int j = 0; j < 4; ++j) {
            float* o = out + (size_t)(m0 + i * 16 + rowadd) * DIMM + n0 + j * 16 + r;
#pragma unroll
            for (int v = 0; v < 8; ++v)
                o[(size_t)v * DIMM] = acc[i][j][v];
        }
    }
}

// ---------------------------------------------------------------------------
// 3) Per-head LayerNorm over D=64 + transpose to [B,H,S,D]. 64 threads/block.
//    blockIdx.x encodes (which, b, s, h)
// ---------------------------------------------------------------------------
__global__ __launch_bounds__(64) void ln_head_kernel(const float* __restrict__ raw,
                                                     const float* __restrict__ gk, const float* __restrict__ bk,
                                                     const float* __restrict__ gv, const float* __restrict__ bv,
                                                     const float* __restrict__ gq, const float* __restrict__ bq,
                                                     float* __restrict__ kb_, float* __restrict__ vb_,
                                                     float* __restrict__ qb_) {
    const int tid = threadIdx.x;
    int id = blockIdx.x;
    const int h = id % NH;          id /= NH;
    const int s = id % SS;          id /= SS;
    const int b = id % NB;          id /= NB;
    const int which = id;           // 0,1,2
    const float* src = raw + (size_t)which * RSZ + ((size_t)b * SS + s) * DIMM + h * DD;
    const float val = src[tid];
    float s1 = val, s2 = val * val;
    for (int o = 16; o >= 1; o >>= 1) { s1 += __shfl_xor(s1, o, 32); s2 += __shfl_xor(s2, o, 32); }
    __shared__ float pa[2], pb[2];
    const int wid = tid >> 5;
    if ((tid & 31) == 0) { pa[wid] = s1; pb[wid] = s2; }
    __syncthreads();
    const float ts  = pa[0] + pa[1];
    const float ts2 = pb[0] + pb[1];
    const float mean = ts * (1.0f / DD);
    const float var  = ts2 * (1.0f / DD) - mean * mean;
    const float inv  = rsqrtf(var + 1e-5f);
    const float* g  = (which == 0) ? gk : ((which == 1) ? gv : gq);
    const float* bb = (which == 0) ? bk : ((which == 1) ? bv : bq);
    float* dst = ((which == 0) ? kb_ : ((which == 1) ? vb_ : qb_)) +
                 (((size_t)b * NH + h) * SS + s) * DD;
    dst[tid] = (val - mean) * inv * g[tid] + bb[tid];
}

// ---------------------------------------------------------------------------
// 4) Sequential scan. One block per (b,h); M/Mom + MLP weights in LDS.
//    pred2 = M@(k/||k||) = (M@k)/||k||, so only two matvecs per step.
// ---------------------------------------------------------------------------
__global__ __launch_bounds__(256) void scan_kernel(const float* __restrict__ kbuf,
                                                   const float* __restrict__ vbuf,
                                                   const float* __restrict__ qbuf,
                                                   const float* __restrict__ M0,
                                                   const float* __restrict__ Mom0,
                                                   const float* __restrict__ w1, const float* __restrict__ b1,
                                                   const float* __restrict__ w2, const float* __restrict__ b2,
                                                   const float* __restrict__ w3, const float* __restrict__ b3,
                                                   const float* __restrict__ lr_scale,
                                                   const float* __restrict__ fg_scale,
                                                   float* __restrict__ attn,
                                                   float* __restrict__ Mout,
                                                   float* __restrict__ MomOut) {
    __shared__ float sM[4096], sMom[4096];
    __shared__ float sW1[192 * 128];
    __shared__ float sW2[128 * 64];
    __shared__ float sW3[128];
    __shared__ float sB1[128], sB2[64], sB3[2];
    __shared__ float kt[64], vt[64], qt[64];
    __shared__ float pred[64], outq[64], sur[64];
    __shared__ float h1[128], h2[64];
    __shared__ float mods[2];
    __shared__ float sInv;

    const int tid = threadIdx.x;
    const int bh  = blockIdx.x;          // b*H + h
    const int h   = bh % NH;
    const int b   = bh / NH;
    const size_t mbase = (size_t)bh * 4096;

    for (int i = tid; i < 4096; i += 256) { sM[i] = M0[mbase + i]; sMom[i] = Mom0[mbase + i]; }
    for (int i = tid; i < 192 * 128; i += 256) sW1[i] = w1[i];
    for (int i = tid; i < 128 * 64;  i += 256) sW2[i] = w2[i];
    if (tid < 128) { sW3[tid] = w3[tid]; sB1[tid] = b1[tid]; }
    if (tid < 64)  sB2[tid] = b2[tid];
    if (tid < 2)   sB3[tid] = b3[tid];
    const float base_lr = sigmoidf_(lr_scale[h]) * 0.2f;   // sigmoid * LR_BASE * 2
    const float base_fg = sigmoidf_(fg_scale[h]) * 0.5f;
    __syncthreads();

    const int drow = tid >> 2;       // 0..63 (4 lanes per row)
    const int seg  = tid & 3;
    const int kss  = seg * 16;
    const size_t tb = (size_t)bh * SS * DD;

    for (int t = 0; t < SS; ++t) {
        const size_t off = tb + (size_t)t * DD;
        if (tid < 64) { kt[tid] = kbuf[off + tid]; vt[tid] = vbuf[off + tid]; qt[tid] = qbuf[off + tid]; }
        __syncthreads();
        if (tid < 32) {
            float s = kt[tid] * kt[tid] + kt[tid + 32] * kt[tid + 32];
            for (int o = 16; o >= 1; o >>= 1) s += __shfl_xor(s, o, 32);
            if (tid == 0) sInv = 1.0f / (sqrtf(s) + 1e-6f);
        }
        // pred = M @ k, outq = M @ q
        float ap = 0.f, aq = 0.f;
        const float* mrow = &sM[drow * 64 + kss];
#pragma unroll
        for (int j = 0; j < 16; ++j) { const float m = mrow[j]; ap += m * kt[kss + j]; aq += m * qt[kss + j]; }
        ap += __shfl_xor(ap, 1, 32); ap += __shfl_xor(ap, 2, 32);
        aq += __shfl_xor(aq, 1, 32); aq += __shfl_xor(aq, 2, 32);
        if (seg == 0) { pred[drow] = ap; outq[drow] = aq; }
        __syncthreads();
        if (tid < 64) {
            sur[tid] = pred[tid] - vt[tid];
            attn[((size_t)b * SS + t) * DIMM + h * DD + tid] = outq[tid];   // [B,S,H*D]
        }
        __syncthreads();
        if (tid < 128) {                       // layer 1: 192 -> 128
            float acc = sB1[tid];
            for (int i = 0; i < 64; ++i) acc += kt[i]  * sW1[i * 128 + tid];
            for (int i = 0; i < 64; ++i) acc += vt[i]  * sW1[(64 + i) * 128 + tid];
            for (int i = 0; i < 64; ++i) acc += sur[i] * sW1[(128 + i) * 128 + tid];
            h1[tid] = geluf_(acc);
        }
        __syncthreads();
        if (tid < 64) {                        // layer 2: 128 -> 64
            float acc = sB2[tid];
            for (int i = 0; i < 128; ++i) acc += h1[i] * sW2[i * 64 + tid];
            h2[tid] = geluf_(acc);
        }
        __syncthreads();
        if (tid < 2) {                         // layer 3: 64 -> 2
            float acc = sB3[tid];
            for (int i = 0; i < 64; ++i) acc += h2[i] * sW3[i * 2 + tid];
            mods[tid] = sigmoidf_(acc);
        }
        __syncthreads();
        const float eff_lr = base_lr * (0.5f + mods[0]);
        const float eff_fg = base_fg * (0.5f + mods[1]);
        const float inv = sInv;
        const float sd = sur[drow];
        const float pf = pred[drow] * inv * eff_fg;      // eff_fg * pred2[d]
#pragma unroll
        for (int j = 0; j < 16; ++j) {
            const int idx = drow * 64 + kss + j;
            const float knk = kt[kss + j] * inv;
            const float mom = 0.9f * sMom[idx] + sd * knk;
            sMom[idx] = mom;
            sM[idx] -= pf * knk + eff_lr * mom;
        }
        __syncthreads();
    }
    for (int i = tid; i < 4096; i += 256) { Mout[mbase + i] = sM[i]; MomOut[mbase + i] = sMom[i]; }
}

// ---------------------------------------------------------------------------
// 5) Elementwise gating: out = sigmoid(gate_raw + gate_b) * out_raw
// ---------------------------------------------------------------------------
__global__ __launch_bounds__(256) void gate_combine_kernel(const float* __restrict__ og,
                                                           const float* __restrict__ gg,
                                                           const float* __restrict__ gb,
                                                           float* __restrict__ out) {
    const size_t i = (size_t)blockIdx.x * 256 + threadIdx.x;
    const int col = (int)(i % DIMM);
    out[i] = sigmoidf_(gg[i] + gb[col]) * og[i];
}

// ---------------------------------------------------------------------------
extern "C" void kernel_launch(void* const* d_in, const int* in_sizes, int n_in,
                              void* d_out, int out_size, void* d_ws, size_t ws_size,
                              hipStream_t stream) {
    const float* x        = (const float*)d_in[0];
    const float* M0       = (const float*)d_in[1];
    const float* Mom0     = (const float*)d_in[2];
    const float* Wk       = (const float*)d_in[3];
    const float* Wv       = (const float*)d_in[4];
    const float* Wq       = (const float*)d_in[5];
    const float* Wo       = (const float*)d_in[6];
    const float* ln_in_g  = (const float*)d_in[7];
    const float* ln_in_b  = (const float*)d_in[8];
    const float* ln_k_g   = (const float*)d_in[9];
    const float* ln_k_b   = (const float*)d_in[10];
    const float* ln_v_g   = (const float*)d_in[11];
    const float* ln_v_b   = (const float*)d_in[12];
    const float* ln_q_g   = (const float*)d_in[13];
    const float* ln_q_b   = (const float*)d_in[14];
    const float* mm_w1    = (const float*)d_in[15];
    const float* mm_b1    = (const float*)d_in[16];
    const float* mm_w2    = (const float*)d_in[17];
    const float* mm_b2    = (const float*)d_in[18];
    const float* mm_w3    = (const float*)d_in[19];
    const float* mm_b3    = (const float*)d_in[20];
    const float* lr_scale = (const float*)d_in[21];
    const float* fg_scale = (const float*)d_in[22];
    const float* gate_w   = (const float*)d_in[23];
    const float* gate_b   = (const float*)d_in[24];

    float* ws   = (float*)d_ws;
    float* xn   = ws;                       // [8192, 768]
    float* raw  = ws + RSZ;                 // 3 * [8192, 768] (consumed by ln_head)
    float* kbuf = ws + 4 * RSZ;             // [B,H,S,D]
    float* vbuf = ws + 5 * RSZ;
    float* qbuf = ws + 6 * RSZ;
    float* attn = ws + RSZ;                 // reuse raw[0] after ln_head (stream-ordered)
    float* og0  = ws + 2 * RSZ;             // reuse raw[1]: attn @ Wo
    float* og1  = ws + 3 * RSZ;             // reuse raw[2]: xn @ gate_w

    float* out0   = (float*)d_out;          // [B,S,DIM]
    float* Mout   = out0 + RSZ;             // [B,H,D,D] = 196608 floats
    float* MomOut = Mout + (size_t)NB * NH * DD * DD;

    ln_in_kernel<<<BS, 256, 0, stream>>>(x, ln_in_g, ln_in_b, xn);

    dim3 ggrid(BS / 64, DIMM / 64);         // (128, 12), 64x64 tile per wave
    gemm768_kernel<<<ggrid, 32, 0, stream>>>(xn, Wk, raw);
    gemm768_kernel<<<ggrid, 32, 0, stream>>>(xn, Wv, raw + RSZ);
    gemm768_kernel<<<ggrid, 32, 0, stream>>>(xn, Wq, raw + 2 * RSZ);

    ln_head_kernel<<<3 * NB * SS * NH, 64, 0, stream>>>(raw, ln_k_g, ln_k_b, ln_v_g, ln_v_b,
                                                        ln_q_g, ln_q_b, kbuf, vbuf, qbuf);

    scan_kernel<<<NB * NH, 256, 0, stream>>>(kbuf, vbuf, qbuf, M0, Mom0,
                                             mm_w1, mm_b1, mm_w2, mm_b2, mm_w3, mm_b3,
                                             lr_scale, fg_scale, attn, Mout, MomOut);

    gemm768_kernel<<<ggrid, 32, 0, stream>>>(attn, Wo, og0);
    gemm768_kernel<<<ggrid, 32, 0, stream>>>(xn, gate_w, og1);
    gate_combine_kernel<<<(unsigned)(RSZ / 256), 256, 0, stream>>>(og0, og1, gate_b, out0);
}